// PolyConv_56255481643507
// MI455X (gfx1250) — compile-verified
//
#include <hip/hip_runtime.h>

// ---------------------------------------------------------------------------
// PolyConv (Chebyshev-like graph filter) for MI455X / gfx1250.
// Scatter-atomic SpMM formulation; memory/atomic bound (no dense GEMM -> no WMMA).
// CDNA5 paths used: global_load_async_to_lds_b32 + s_wait_asynccnt (edge-index
// staging), ds broadcast reads, global_prefetch_b8 (row prefetch),
// global_atomic_add_f32 (L2-resident scatter).
// d_ws layout: [ dinv: nn4 floats | featc: N*64 floats | agg: N*64 floats ]
// ---------------------------------------------------------------------------

#define WAVE 32
#define TB   256
#define WPB  (TB / WAVE)   // 8 waves per block

__global__ void k_zero_deg(float* __restrict__ deg, int n) {
    int i = blockIdx.x * blockDim.x + threadIdx.x;
    if (i < n) deg[i] = 0.0f;
}

__global__ void k_degree(const int* __restrict__ dst, float* __restrict__ deg, int n_edges) {
    int e = blockIdx.x * blockDim.x + threadIdx.x;
    if (e < n_edges) atomicAdd(&deg[dst[e]], 1.0f);
}

__global__ void k_dinv(float* __restrict__ deg, int n) {
    int i = blockIdx.x * blockDim.x + threadIdx.x;
    if (i < n) {
        float d = deg[i];
        deg[i] = 1.0f / sqrtf(fmaxf(d, 1.0f));   // in-place: deg -> d^{-1/2}
    }
}

// featc = feat; h = theta[0]*feat; agg = 0   (float4-vectorized over N*64 elems)
__global__ void k_init(const float4* __restrict__ feat, const float* __restrict__ theta,
                       float4* __restrict__ featc, float4* __restrict__ h,
                       float4* __restrict__ agg, int n4) {
    int i = blockIdx.x * blockDim.x + threadIdx.x;
    if (i >= n4) return;
    float t0 = theta[0];                 // scalar broadcast load
    float4 f = feat[i];
    featc[i] = f;
    float4 hh; hh.x = t0 * f.x; hh.y = t0 * f.y; hh.z = t0 * f.z; hh.w = t0 * f.w;
    h[i] = hh;
    float4 z; z.x = 0.f; z.y = 0.f; z.z = 0.f; z.w = 0.f;
    agg[i] = z;
}

// One wave32 per 32-edge chunk. Lane l async-stages src[base+l], dst[base+l]
// into this wave's LDS slots (2 coalesced async ops per chunk instead of 2
// broadcast VMEM loads per edge). Then for each edge j: broadcast-read (s,d)
// from LDS, prefetch edge j+1's feature row, gather 32 lanes x float2 of
// featc[s], and scatter agg[d] += featc[s]*dinv[s] via global_atomic_add_f32.
__global__ void k_scatter(const float* __restrict__ featc, const float* __restrict__ dinv,
                          const int* __restrict__ src, const int* __restrict__ dst,
                          float* __restrict__ agg, int n_edges) {
    __shared__ int sbuf[WPB * 2 * WAVE];              // per-wave: 32 src | 32 dst
    const int lane = threadIdx.x & (WAVE - 1);
    const int wib  = threadIdx.x >> 5;                // wave index in block
    const int wgl  = blockIdx.x * WPB + wib;          // global wave id
    const int nwv  = gridDim.x * WPB;                 // total waves

    int* ws = &sbuf[wib * 2 * WAVE];                  // this wave's LDS slots
    // Low 32 bits of a generic LDS pointer == LDS byte offset (ISA aperture rule);
    // both DS ops and async-to-LDS add the wave's LDS_BASE, so this is consistent.
    const unsigned lds_src = (unsigned)(uintptr_t)(void*)&ws[0];
    const unsigned lds_dst = (unsigned)(uintptr_t)(void*)&ws[WAVE];

    for (long long base = (long long)wgl * WAVE; base < n_edges;
         base += (long long)nwv * WAVE) {
        int e = (int)base + lane;
        if (e > n_edges - 1) e = n_edges - 1;         // clamp tail (stays in-bounds)

        // ---- ASYNCcnt-tracked coalesced index staging into LDS ----
        unsigned long long ga_s = (unsigned long long)(uintptr_t)(src + e);
        unsigned long long ga_d = (unsigned long long)(uintptr_t)(dst + e);
        unsigned la_s = lds_src + (unsigned)lane * 4u;
        unsigned la_d = lds_dst + (unsigned)lane * 4u;
        asm volatile("global_load_async_to_lds_b32 %0, %1, off"
                     :: "v"(la_s), "v"(ga_s) : "memory");
        asm volatile("global_load_async_to_lds_b32 %0, %1, off"
                     :: "v"(la_d), "v"(ga_d) : "memory");
        asm volatile("s_wait_asynccnt 0x0" ::: "memory");

        int cnt = n_edges - (int)base;
        if (cnt > WAVE) cnt = WAVE;

        int s = ws[0];                                // LDS broadcast read
        for (int j = 0; j < cnt; ++j) {
            int d  = ws[WAVE + j];                    // LDS broadcast read
            int sn = (j + 1 < cnt) ? ws[j + 1] : s;
            // Prefetch next edge's 256B feature row while this edge's atomics drain
            __builtin_prefetch(featc + (size_t)sn * 64 + (size_t)lane * 2, 0, 0);

            float w = dinv[s];
            const float2 v = *(const float2*)(featc + (size_t)s * 64 + (size_t)lane * 2);
            float* o = agg + (size_t)d * 64 + (size_t)lane * 2;
            atomicAdd(o,     v.x * w);
            atomicAdd(o + 1, v.y * w);
            s = sn;
        }
    }
}

// featc = featc - agg*dinv[node]; h += theta[k]*featc; agg = 0 (for next iter)
__global__ void k_update(float4* __restrict__ featc, float4* __restrict__ agg,
                         const float* __restrict__ dinv, float4* __restrict__ h,
                         const float* __restrict__ theta, int k, int n4) {
    int i = blockIdx.x * blockDim.x + threadIdx.x;
    if (i >= n4) return;
    float th = theta[k];
    float w  = dinv[i >> 4];             // 16 float4 per 64-wide node row
    float4 a = agg[i];
    float4 f = featc[i];
    f.x -= a.x * w; f.y -= a.y * w; f.z -= a.z * w; f.w -= a.w * w;
    featc[i] = f;
    float4 hh = h[i];
    hh.x += th * f.x; hh.y += th * f.y; hh.z += th * f.z; hh.w += th * f.w;
    h[i] = hh;
    float4 z; z.x = 0.f; z.y = 0.f; z.z = 0.f; z.w = 0.f;
    agg[i] = z;
}

extern "C" void kernel_launch(void* const* d_in, const int* in_sizes, int n_in,
                              void* d_out, int out_size, void* d_ws, size_t ws_size,
                              hipStream_t stream) {
    const float* feat  = (const float*)d_in[0];
    const float* theta = (const float*)d_in[1];
    const int*   src   = (const int*)d_in[2];
    const int*   dst   = (const int*)d_in[3];
    float*       h     = (float*)d_out;

    const int n_nodes = in_sizes[0] / 64;
    const int n_theta = in_sizes[1];
    const int n_edges = in_sizes[2];

    const int nn4 = (n_nodes + 3) & ~3;          // keep featc 16B-aligned
    float* dinv  = (float*)d_ws;
    float* featc = dinv + nn4;
    float* agg   = featc + (size_t)n_nodes * 64;

    const int n4 = n_nodes * 16;                 // float4 count of N x 64

    // degree -> d^{-1/2}
    k_zero_deg<<<(n_nodes + TB - 1) / TB, TB, 0, stream>>>(dinv, n_nodes);
    k_degree  <<<(n_edges + TB - 1) / TB, TB, 0, stream>>>(dst, dinv, n_edges);
    k_dinv    <<<(n_nodes + TB - 1) / TB, TB, 0, stream>>>(dinv, n_nodes);

    // h = theta0*feat; featc = feat; agg = 0
    k_init<<<(n4 + TB - 1) / TB, TB, 0, stream>>>((const float4*)feat, theta,
                                                  (float4*)featc, (float4*)h,
                                                  (float4*)agg, n4);

    // one wave per 32-edge chunk
    long long chunks    = ((long long)n_edges + WAVE - 1) / WAVE;
    int       blocks_sc = (int)((chunks + WPB - 1) / WPB);
    if (blocks_sc < 1) blocks_sc = 1;

    for (int k = 1; k < n_theta; ++k) {
        k_scatter<<<blocks_sc, TB, 0, stream>>>(featc, dinv, src, dst, agg, n_edges);
        k_update <<<(n4 + TB - 1) / TB, TB, 0, stream>>>((float4*)featc, (float4*)agg,
                                                         dinv, (float4*)h, theta, k, n4);
    }
}